// AdjSAGE_23596550324897
// MI455X (gfx1250) — compile-verified
//
#include <hip/hip_runtime.h>
#include <math.h>

#define NN   100000
#define EE   1600000
#define DF   128
#define DOUT 40
#define LDW  (DF + 4)   // LDS row stride (floats): 132 -> conflict-free column reads, 16B-aligned rows

typedef __attribute__((ext_vector_type(2))) float v2f;
typedef __attribute__((ext_vector_type(4))) float v4f;
typedef __attribute__((ext_vector_type(8))) float v8f;

// gfx1250 async global->LDS copy (ASYNCcnt-tracked, no VGPR round-trip).
// LDS byte address = low 32 bits of the generic shared pointer.
__device__ __forceinline__ void async_cp16(void* lds_dst, const float* gsrc) {
  unsigned int la = (unsigned int)(uintptr_t)lds_dst;
  asm volatile("global_load_async_to_lds_b128 %0, %1, off"
               :: "v"(la), "v"((unsigned long long)(uintptr_t)gsrc)
               : "memory");
}
__device__ __forceinline__ void async_wait0() {
  asm volatile("s_wait_asynccnt 0" ::: "memory");
}

// ---------------- zero (float4 stores) ----------------
__global__ void k_zero4(float* __restrict__ p, long n) {
  long i = ((long)blockIdx.x * blockDim.x + threadIdx.x) * 4;
  if (i < n) {
    v4f z = {};
    *(v4f*)(p + i) = z;
  }
}

// ---------------- degree + inverse ----------------
__global__ void k_degree(const int* __restrict__ edst, float* __restrict__ deg) {
  int e = blockIdx.x * blockDim.x + threadIdx.x;
  if (e < EE) atomicAdd(&deg[edst[e]], 1.0f);
}

__global__ void k_invdeg(float* __restrict__ deg) {
  int i = blockIdx.x * blockDim.x + threadIdx.x;
  if (i < NN) deg[i] = 1.0f / fmaxf(deg[i], 1.0f);
}

// ---------------- edge aggregation: one wave32 per edge, float4 per lane ----------------
__global__ __launch_bounds__(256)
void k_aggregate(const float* __restrict__ feat, const int* __restrict__ esrc,
                 const int* __restrict__ edst, float* __restrict__ agg) {
  int t    = blockIdx.x * blockDim.x + threadIdx.x;
  int e    = t >> 5;
  int lane = t & 31;
  if (e >= EE) return;
  if (lane == 0 && (e + 2048) < EE) {           // gfx1250 global_prefetch_b8
    __builtin_prefetch(esrc + e + 2048, 0, 0);
    __builtin_prefetch(edst + e + 2048, 0, 0);
  }
  int s = esrc[e];
  int d = edst[e];
  v4f v = *(const v4f*)(feat + (size_t)s * DF + lane * 4);
  float* ap = agg + (size_t)d * DF + lane * 4;
  atomicAdd(ap + 0, v.x);
  atomicAdd(ap + 1, v.y);
  atomicAdd(ap + 2, v.z);
  atomicAdd(ap + 3, v.w);
}

// ---------------- SAGE dense layer via V_WMMA_F32_16X16X4_F32 ----------------
// out[16, dout] = (inv_deg .* agg) @ Wl^T + xin @ Wr^T + bias  (optional relu)
// Block: 256 threads = 8 waves; wave w owns 16-col tiles w, w+8, ...
// Two accumulator chains (agg / root); inv_deg applied in epilogue.
// Weight-row pointers are clamped instead of masked: column n of D depends only on
// column n of B, so out-of-range columns hold garbage that is never stored.
__global__ __launch_bounds__(256)
void k_sage_gemm(const float* __restrict__ agg, const float* __restrict__ xin,
                 const float* __restrict__ invdeg,
                 const float* __restrict__ Wl, const float* __restrict__ Wr,
                 const float* __restrict__ bias,
                 float* __restrict__ out, int dout, int do_relu) {
  __shared__ float As[16][LDW];
  __shared__ float Xs[16][LDW];
  __shared__ float sInv[16];

  const int row0 = blockIdx.x * 16;
  const int tid  = threadIdx.x;

  if (tid < 16) sInv[tid] = invdeg[row0 + tid];

  // Async-stage both 16x128 activation tiles into LDS (512 x 16B chunks each, 2 per thread).
  for (int i = tid; i < 16 * (DF / 4); i += 256) {
    int r  = i >> 5;
    int c4 = i & 31;
    const size_t g = (size_t)(row0 + r) * DF + c4 * 4;
    async_cp16(&As[r][c4 * 4], agg + g);
    async_cp16(&Xs[r][c4 * 4], xin + g);
  }
  async_wait0();
  __syncthreads();

  const int wave  = tid >> 5;
  const int lane  = tid & 31;
  const int m     = lane & 15;           // A row / B&C column index
  const int khalf = (lane >> 4) << 1;    // K sub-offset: 0 (lanes 0-15) or 2 (lanes 16-31)
  const int mhi   = (lane >> 4) << 3;    // C row base: 0 (lo lanes) or 8 (hi lanes)
  const int ntiles = (dout + 15) >> 4;

  for (int tIdx = wave; tIdx < ntiles; tIdx += 8) {
    const int colbase = tIdx << 4;
    const int n  = colbase + m;
    const int nc = (n < dout) ? n : (dout - 1);      // clamp: no masking needed
    const float* wl = Wl + (size_t)nc * DF;
    const float* wr = Wr + (size_t)nc * DF;

    v8f cA = {};
    v8f cX = {};
#pragma unroll
    for (int k0 = 0; k0 < DF; k0 += 4) {
      v2f aA = *(const v2f*)&As[m][k0 + khalf];
      v2f aX = *(const v2f*)&Xs[m][k0 + khalf];
      v2f bL = *(const v2f*)(wl + k0 + khalf);
      v2f bR = *(const v2f*)(wr + k0 + khalf);
      cA = __builtin_amdgcn_wmma_f32_16x16x4_f32(false, aA, false, bL, (short)0, cA, false, false);
      cX = __builtin_amdgcn_wmma_f32_16x16x4_f32(false, aX, false, bR, (short)0, cX, false, false);
    }

    if (n < dout) {
      float bv = bias[n];
#pragma unroll
      for (int r = 0; r < 8; ++r) {
        float v = cA[r] * sInv[mhi + r] + cX[r] + bv;
        if (do_relu) v = fmaxf(v, 0.0f);
        out[(size_t)(row0 + mhi + r) * dout + n] = v;
      }
    }
  }
}

// ---------------- log_softmax over 40 classes, one wave32 per row, in place ----------------
__global__ __launch_bounds__(256)
void k_logsoftmax(float* __restrict__ out) {
  int t    = blockIdx.x * blockDim.x + threadIdx.x;
  int row  = t >> 5;
  int lane = t & 31;
  if (row >= NN) return;
  float* p = out + (size_t)row * DOUT;
  float v0 = p[lane];                                   // lane < 32 < 40: always valid
  float v1 = (lane + 32 < DOUT) ? p[lane + 32] : -INFINITY;
  float mx = fmaxf(v0, v1);
#pragma unroll
  for (int o = 16; o > 0; o >>= 1) mx = fmaxf(mx, __shfl_xor(mx, o, 32));
  float s = expf(v0 - mx) + ((lane + 32 < DOUT) ? expf(v1 - mx) : 0.0f);
#pragma unroll
  for (int o = 16; o > 0; o >>= 1) s += __shfl_xor(s, o, 32);
  float lse = mx + logf(s);
  p[lane] = v0 - lse;
  if (lane + 32 < DOUT) p[lane + 32] = v1 - lse;
}

// ---------------- driver ----------------
extern "C" void kernel_launch(void* const* d_in, const int* in_sizes, int n_in,
                              void* d_out, int out_size, void* d_ws, size_t ws_size,
                              hipStream_t stream) {
  const float* x    = (const float*)d_in[0];
  const int*   esrc = (const int*)d_in[1];
  const int*   edst = (const int*)d_in[2];
  const float* Wl0  = (const float*)d_in[3];
  const float* Wr0  = (const float*)d_in[4];
  const float* b0   = (const float*)d_in[5];
  const float* Wl1  = (const float*)d_in[6];
  const float* Wr1  = (const float*)d_in[7];
  const float* b1   = (const float*)d_in[8];
  const float* Wl2  = (const float*)d_in[9];
  const float* Wr2  = (const float*)d_in[10];
  const float* b2   = (const float*)d_in[11];
  float* out = (float*)d_out;

  // Workspace layout: inv_deg[N] | agg[N*128] | h[N*128]  (~103 MB)
  char* ws = (char*)d_ws;
  float* invdeg = (float*)ws;
  size_t off = ((size_t)NN * 4 + 511) & ~(size_t)511;
  float* agg = (float*)(ws + off);
  off += (size_t)NN * DF * 4;
  float* h = (float*)(ws + off);

  const int T = 256;
  const long aggN = (long)NN * DF;
  const int gZeroN   = (int)((NN / 4 + T - 1) / T);
  const int gZeroAgg = (int)((aggN / 4 + T - 1) / T);
  const int gN       = (NN + T - 1) / T;
  const int gEdge    = (EE + T - 1) / T;
  const int gAggr    = (int)(((long)EE * 32 + T - 1) / T);
  const int gGemm    = NN / 16;                 // 100000 % 16 == 0
  const int gRowWave = (int)(((long)NN * 32 + T - 1) / T);

  // degree -> inv_deg
  k_zero4 <<<gZeroN, T, 0, stream>>>(invdeg, NN);
  k_degree<<<gEdge,  T, 0, stream>>>(edst, invdeg);
  k_invdeg<<<gN,     T, 0, stream>>>(invdeg);

  // layer 0
  k_zero4    <<<gZeroAgg, T, 0, stream>>>(agg, aggN);
  k_aggregate<<<gAggr,    T, 0, stream>>>(x, esrc, edst, agg);
  k_sage_gemm<<<gGemm,    T, 0, stream>>>(agg, x, invdeg, Wl0, Wr0, b0, h, DF, 1);

  // layer 1 (h updated in place: each block stages its own 16 rows into LDS first)
  k_zero4    <<<gZeroAgg, T, 0, stream>>>(agg, aggN);
  k_aggregate<<<gAggr,    T, 0, stream>>>(h, esrc, edst, agg);
  k_sage_gemm<<<gGemm,    T, 0, stream>>>(agg, h, invdeg, Wl1, Wr1, b1, h, DF, 1);

  // layer 2 -> logits in d_out, then in-place log_softmax
  k_zero4    <<<gZeroAgg, T, 0, stream>>>(agg, aggN);
  k_aggregate<<<gAggr,    T, 0, stream>>>(h, esrc, edst, agg);
  k_sage_gemm<<<gGemm,    T, 0, stream>>>(agg, h, invdeg, Wl2, Wr2, b2, out, DOUT, 0);
  k_logsoftmax<<<gRowWave, T, 0, stream>>>(out);
}